// Three_Dimensional_LUT_3126736192196
// MI455X (gfx1250) — compile-verified
//
#include <hip/hip_runtime.h>

// Trilinear 3D-LUT apply for img (8,3,1024,1024) f32 with LUT (8,3,33,33,33) f32.
// Strategy:
//   1) repack LUT to channel-interleaved float4 entries in d_ws:
//      LUTp[b][n] = {LUT[b,0,n], LUT[b,1,n], LUT[b,2,n], 0}, n = (z*33+y)*33+x
//      -> 8 aligned b128 gathers per pixel instead of 24 scalar gathers.
//      Repacked LUT = 4.6 MB total -> fully L2-resident (192 MB L2).
//   2) main kernel: 4 pixels/thread, NT streaming loads/stores for img/out,
//      global_prefetch for the stream, gathers from L2-resident packed LUT.

typedef float v4f __attribute__((ext_vector_type(4)));

#define LUT_N   33
#define LUT_N3  (LUT_N * LUT_N * LUT_N)   /* 35937 */
#define BATCH   8
#define PLANE   (1024 * 1024)
#define PF_DIST 8192                       /* floats ahead (~32 KB) */

__device__ __forceinline__ v4f lerp4(v4f a, v4f b, float t) {
    // matches reference: a*(1-t) + b*t
    return a * (1.0f - t) + b * t;
}

__global__ __launch_bounds__(256) void lut_repack_kernel(
    const float* __restrict__ lut, v4f* __restrict__ lutp)
{
    int t = blockIdx.x * blockDim.x + threadIdx.x;
    if (t >= BATCH * LUT_N3) return;
    int b = t / LUT_N3;
    int n = t - b * LUT_N3;
    const float* base = lut + (size_t)b * 3 * LUT_N3;
    v4f v;
    v.x = base[n];
    v.y = base[LUT_N3 + n];
    v.z = base[2 * LUT_N3 + n];
    v.w = 0.0f;
    lutp[t] = v;
}

__global__ __launch_bounds__(256) void lut_apply_packed(
    const float* __restrict__ img, const v4f* __restrict__ lutp,
    float* __restrict__ out)
{
    const int b  = blockIdx.y;
    const int p0 = (blockIdx.x * blockDim.x + threadIdx.x) * 4;
    if (p0 >= PLANE) return;

    const float* xs = img + (size_t)b * 3 * PLANE;
    const float* ys = xs + PLANE;
    const float* zs = ys + PLANE;

    if (p0 + PF_DIST + 4 <= PLANE) {
        __builtin_prefetch(xs + p0 + PF_DIST, 0, 3);
        __builtin_prefetch(ys + p0 + PF_DIST, 0, 3);
        __builtin_prefetch(zs + p0 + PF_DIST, 0, 3);
    }

    // Non-temporal streaming loads: don't let the 200 MB stream evict the
    // L2-resident packed LUT.
    v4f xv = __builtin_nontemporal_load((const v4f*)(xs + p0));
    v4f yv = __builtin_nontemporal_load((const v4f*)(ys + p0));
    v4f zv = __builtin_nontemporal_load((const v4f*)(zs + p0));

    const v4f* __restrict__ L = lutp + (size_t)b * LUT_N3;

    v4f res[4];
#pragma unroll
    for (int j = 0; j < 4; ++j) {
        float px = fminf(fmaxf(xv[j] * 32.0f, 0.0f), 32.0f);
        float py = fminf(fmaxf(yv[j] * 32.0f, 0.0f), 32.0f);
        float pz = fminf(fmaxf(zv[j] * 32.0f, 0.0f), 32.0f);
        float fx = floorf(px), fy = floorf(py), fz = floorf(pz);
        int x0 = (int)fx, y0 = (int)fy, z0 = (int)fz;
        int x1 = min(x0 + 1, LUT_N - 1);
        int y1 = min(y0 + 1, LUT_N - 1);
        int z1 = min(z0 + 1, LUT_N - 1);
        float wx = px - fx, wy = py - fy, wz = pz - fz;

        int zy00 = (z0 * LUT_N + y0) * LUT_N;
        int zy01 = (z0 * LUT_N + y1) * LUT_N;
        int zy10 = (z1 * LUT_N + y0) * LUT_N;
        int zy11 = (z1 * LUT_N + y1) * LUT_N;

        // 8 x b128 gathers (all three channels per corner in one load)
        v4f c000 = L[zy00 + x0], c001 = L[zy00 + x1];
        v4f c010 = L[zy01 + x0], c011 = L[zy01 + x1];
        v4f c100 = L[zy10 + x0], c101 = L[zy10 + x1];
        v4f c110 = L[zy11 + x0], c111 = L[zy11 + x1];

        v4f c00 = lerp4(c000, c001, wx);
        v4f c01 = lerp4(c010, c011, wx);
        v4f c10 = lerp4(c100, c101, wx);
        v4f c11 = lerp4(c110, c111, wx);
        v4f c0  = lerp4(c00, c01, wy);
        v4f c1  = lerp4(c10, c11, wy);
        res[j]  = lerp4(c0, c1, wz);
    }

    // Transpose 4 pixel results into per-channel b128 stores.
    v4f r0 = { res[0].x, res[1].x, res[2].x, res[3].x };
    v4f r1 = { res[0].y, res[1].y, res[2].y, res[3].y };
    v4f r2 = { res[0].z, res[1].z, res[2].z, res[3].z };

    float* ob = out + (size_t)b * 3 * PLANE + p0;
    __builtin_nontemporal_store(r0, (v4f*)(ob));
    __builtin_nontemporal_store(r1, (v4f*)(ob + PLANE));
    __builtin_nontemporal_store(r2, (v4f*)(ob + 2 * PLANE));
}

// Fallback if the workspace is too small for the repacked LUT: gather the
// three channels separately from the original layout.
__global__ __launch_bounds__(256) void lut_apply_direct(
    const float* __restrict__ img, const float* __restrict__ lut,
    float* __restrict__ out)
{
    const int b = blockIdx.y;
    const int p = blockIdx.x * blockDim.x + threadIdx.x;
    if (p >= PLANE) return;

    const float* xs = img + (size_t)b * 3 * PLANE;
    float px = fminf(fmaxf(xs[p] * 32.0f, 0.0f), 32.0f);
    float py = fminf(fmaxf(xs[PLANE + p] * 32.0f, 0.0f), 32.0f);
    float pz = fminf(fmaxf(xs[2 * PLANE + p] * 32.0f, 0.0f), 32.0f);
    float fx = floorf(px), fy = floorf(py), fz = floorf(pz);
    int x0 = (int)fx, y0 = (int)fy, z0 = (int)fz;
    int x1 = min(x0 + 1, LUT_N - 1);
    int y1 = min(y0 + 1, LUT_N - 1);
    int z1 = min(z0 + 1, LUT_N - 1);
    float wx = px - fx, wy = py - fy, wz = pz - fz;

    int i000 = (z0 * LUT_N + y0) * LUT_N + x0;
    int i001 = (z0 * LUT_N + y0) * LUT_N + x1;
    int i010 = (z0 * LUT_N + y1) * LUT_N + x0;
    int i011 = (z0 * LUT_N + y1) * LUT_N + x1;
    int i100 = (z1 * LUT_N + y0) * LUT_N + x0;
    int i101 = (z1 * LUT_N + y0) * LUT_N + x1;
    int i110 = (z1 * LUT_N + y1) * LUT_N + x0;
    int i111 = (z1 * LUT_N + y1) * LUT_N + x1;

#pragma unroll
    for (int c = 0; c < 3; ++c) {
        const float* Lc = lut + ((size_t)b * 3 + c) * LUT_N3;
        float c00 = Lc[i000] * (1.0f - wx) + Lc[i001] * wx;
        float c01 = Lc[i010] * (1.0f - wx) + Lc[i011] * wx;
        float c10 = Lc[i100] * (1.0f - wx) + Lc[i101] * wx;
        float c11 = Lc[i110] * (1.0f - wx) + Lc[i111] * wx;
        float c0  = c00 * (1.0f - wy) + c01 * wy;
        float c1  = c10 * (1.0f - wy) + c11 * wy;
        out[((size_t)b * 3 + c) * PLANE + p] = c0 * (1.0f - wz) + c1 * wz;
    }
}

extern "C" void kernel_launch(void* const* d_in, const int* in_sizes, int n_in,
                              void* d_out, int out_size, void* d_ws, size_t ws_size,
                              hipStream_t stream) {
    (void)in_sizes; (void)n_in; (void)out_size;
    const float* img = (const float*)d_in[0];
    const float* lut = (const float*)d_in[1];
    float* out = (float*)d_out;

    const size_t ws_need = (size_t)BATCH * LUT_N3 * sizeof(v4f); // ~4.6 MB

    dim3 grid4(PLANE / 4 / 256, BATCH); // 1024 x 8 blocks, 4 px/thread

    if (ws_size >= ws_need) {
        v4f* lutp = (v4f*)d_ws;
        int n_entries = BATCH * LUT_N3;
        lut_repack_kernel<<<(n_entries + 255) / 256, 256, 0, stream>>>(lut, lutp);
        lut_apply_packed<<<grid4, 256, 0, stream>>>(img, lutp, out);
    } else {
        dim3 grid1(PLANE / 256, BATCH);
        lut_apply_direct<<<grid1, 256, 0, stream>>>(img, lut, out);
    }
}